// ObservationEncoder_77644418777368
// MI455X (gfx1250) — compile-verified
//
#include <hip/hip_runtime.h>

typedef __attribute__((ext_vector_type(2))) float v2f;
typedef __attribute__((ext_vector_type(8))) float v8f;

#define NB 16
#define NWAVE 8

// ---- LDS layout (float offsets) ----
#define OFF_MS    0        // moveset sums:       288 rows x 16
#define OFF_DMIA  4608     // dmia input:         288 rows x 52
#define OFF_PUB   19584    // enc_public input:   192 rows x 72
#define OFF_OURA  33408    // our_active input:    16 rows x 256
#define OFF_OURB  37504    // our_bench input:     80 rows x 192
#define OFF_OPPA  52864    // opp_active input:    16 rows x 160
#define OFF_OPPB  55424    // opp_bench input:     80 rows x 96
#define OFF_BV    63104    // emb_bv:              32 rows x 48
#define OFF_FIN   64640    // field input:         32 rows x 24
#define OFF_FLD   65408    // field output:        32 rows x 32
#define LDS_FLOATS 66432
#define LDS_BYTES (LDS_FLOATS * 4)

struct Args {
  const int   *__restrict__ public_dex, *__restrict__ private_dex;
  const int   *__restrict__ public_moves, *__restrict__ private_moves, *__restrict__ active_moves;
  const int   *__restrict__ public_item, *__restrict__ private_item;
  const int   *__restrict__ public_ability, *__restrict__ private_ability;
  const float *__restrict__ public_hp;
  const int   *__restrict__ public_status;
  const float *__restrict__ private_stats, *__restrict__ weather, *__restrict__ slot, *__restrict__ abv;
  const float *__restrict__ dex_tab, *__restrict__ move_id_tab, *__restrict__ move_pp_tab;
  const float *__restrict__ item_tab, *__restrict__ ability_tab, *__restrict__ status_tab;
  const float *__restrict__ W_move, *__restrict__ b_move, *__restrict__ W_moveset;
  const float *__restrict__ W_hp, *__restrict__ W_stat, *__restrict__ W_dmia;
  const float *__restrict__ W_weather, *__restrict__ b_weather;
  const float *__restrict__ W_slot, *__restrict__ W_boosts, *__restrict__ W_public;
  const float *__restrict__ W_active, *__restrict__ W_field;
  const float *__restrict__ W_oura, *__restrict__ b_oura, *__restrict__ W_ourb, *__restrict__ b_ourb;
  const float *__restrict__ W_oppa, *__restrict__ b_oppa, *__restrict__ W_oppb, *__restrict__ b_oppb;
  float *__restrict__ out;
};

__device__ __forceinline__ v8f wmma4(v2f a, v2f b, v8f c) {
  // D = A(16x4 f32) * B(4x16 f32) + C(16x16 f32), exact fp32 semantics.
  return __builtin_amdgcn_wmma_f32_16x16x4_f32(false, a, false, b, (short)0, c, false, false);
}

// Flat tile-list GEMM: wave-strided over (rowTile, colTile) pairs.
// fa(row, k0) -> {A[row][k0], A[row][k0+1]};  fb(col, k0) -> {B[k0][col], B[k0+1][col]}
// (lane L supplies row/col L&15 at k-offsets {0,1} lanes 0-15 / {2,3} lanes 16-31)
template <int nRT, int nCT, int nKS, typename FA, typename FB, typename FD>
__device__ __forceinline__ void run_gemm(FA fa, FB fb, FD fd) {
  const int wave = threadIdx.x >> 5;
  const int ln   = threadIdx.x & 31;
  const int l15  = ln & 15;
  const int kadd = (ln & 16) ? 2 : 0;
  for (int t = wave; t < nRT * nCT; t += NWAVE) {
    const int rt = t / nCT, ct = t % nCT;
    v8f acc = {0.f, 0.f, 0.f, 0.f, 0.f, 0.f, 0.f, 0.f};
    for (int ks = 0; ks < nKS; ++ks) {
      const int k0 = ks * 4 + kadd;
      acc = wmma4(fa(rt * 16 + l15, k0), fb(ct * 16 + l15, k0), acc);
    }
    fd(rt, ct, acc);
  }
}

// Row-blocked GEMM: wave-strided over row tiles; all NCT column-tile accumulators
// held in registers so each A fragment is loaded from LDS exactly once and feeds
// NCT back-to-back WMMAs. Use when nRT >= NWAVE so all waves stay busy.
template <int NCT, int nKS, typename FA, typename FB, typename FD>
__device__ __forceinline__ void run_gemm_rb(int nRT, FA fa, FB fb, FD fd) {
  const int wave = threadIdx.x >> 5;
  const int ln   = threadIdx.x & 31;
  const int l15  = ln & 15;
  const int kadd = (ln & 16) ? 2 : 0;
  for (int rt = wave; rt < nRT; rt += NWAVE) {
    v8f acc[NCT];
#pragma unroll
    for (int c = 0; c < NCT; ++c) acc[c] = (v8f){0.f, 0.f, 0.f, 0.f, 0.f, 0.f, 0.f, 0.f};
    for (int ks = 0; ks < nKS; ++ks) {
      const int k0 = ks * 4 + kadd;
      const v2f av = fa(rt * 16 + l15, k0);
#pragma unroll
      for (int c = 0; c < NCT; ++c)
        acc[c] = wmma4(av, fb(c * 16 + l15, k0), acc[c]);
    }
#pragma unroll
    for (int c = 0; c < NCT; ++c) fd(rt, c, acc[c]);
  }
}

// Column-blocked GEMM (executed by ONE wave for one column tile ct): each weight
// B fragment is loaded from global exactly once and feeds all NRT row-tile
// accumulators -- cuts B-fragment L2 traffic by NRT for big streaming weights.
template <int NRT, int nKS, typename FA, typename FB, typename FD>
__device__ __forceinline__ void gemm_colblk(int ct, FA fa, FB fb, FD fd) {
  const int ln   = threadIdx.x & 31;
  const int l15  = ln & 15;
  const int kadd = (ln & 16) ? 2 : 0;
  v8f acc[NRT];
#pragma unroll
  for (int r = 0; r < NRT; ++r) acc[r] = (v8f){0.f, 0.f, 0.f, 0.f, 0.f, 0.f, 0.f, 0.f};
  for (int ks = 0; ks < nKS; ++ks) {
    const int k0 = ks * 4 + kadd;
    const v2f bv = fb(ct * 16 + l15, k0);
#pragma unroll
    for (int r = 0; r < NRT; ++r)
      acc[r] = wmma4(fa(r * 16 + l15, k0), bv, acc[r]);
  }
#pragma unroll
  for (int r = 0; r < NRT; ++r) fd(r, ct, acc[r]);
}

__global__ void __launch_bounds__(256)
obs_encoder_kernel(Args A, int Btot) {
  extern __shared__ float smem[];
  float* sm = smem;
  const int b0   = blockIdx.x * NB;
  const int tid  = threadIdx.x;
  const int ln   = tid & 31;
  const int l15  = ln & 15;
  const int half = (ln >> 4) & 1;

  const long long OUT_OURB  = (long long)Btot * 128;
  const long long OUT_OPPA  = (long long)Btot * 448;
  const long long OUT_OPPB  = (long long)Btot * 576;
  const long long OUT_MOVES = (long long)Btot * 896;

  // ============ Phase 0: move encoder (relu(X@W_move.T+b)), moveset sums, moves out ============
  // Rows: (batch, entity(19), move(8)) flattened; tile = 2 entities (16 rows).
  {
    const int wave = tid >> 5;
    const int kadd = half * 2;
    for (int t = wave; t < (NB * 19) / 2; t += NWAVE) {   // 152 tiles
      const int E = 2 * t + (l15 >> 3);                   // entity for this lane's A-row
      const int m = l15 & 7;
      const int b = E / 19, e = E % 19;
      const int* mp;
      if (e < 12)      mp = A.public_moves  + (((b0 + b) * 12 + e) * 8 + m) * 2;
      else if (e < 18) mp = A.private_moves + (((b0 + b) * 6 + (e - 12)) * 8 + m) * 2;
      else             mp = A.active_moves  + ((b0 + b) * 8 + m) * 2;
      const int id = mp[0], pp = mp[1];
      v8f acc = {0.f, 0.f, 0.f, 0.f, 0.f, 0.f, 0.f, 0.f};
#pragma unroll
      for (int ks = 0; ks < 3; ++ks) {                    // K = 12
        const int k0 = ks * 4 + kadd;
        v2f a, w;
        a.x = (k0     < 8) ? A.move_id_tab[id * 8 + k0]     : A.move_pp_tab[pp * 4 + k0 - 8];
        a.y = (k0 + 1 < 8) ? A.move_id_tab[id * 8 + k0 + 1] : A.move_pp_tab[pp * 4 + k0 - 7];
        w.x = A.W_move[l15 * 12 + k0];
        w.y = A.W_move[l15 * 12 + k0 + 1];
        acc = wmma4(a, w, acc);
      }
      const float bias = A.b_move[l15];
#pragma unroll
      for (int v = 0; v < 8; ++v) acc[v] = fmaxf(acc[v] + bias, 0.f);
      // D layout: lanes 0-15 hold entity 2t (rows 0-7), lanes 16-31 entity 2t+1 (rows 8-15);
      // VGPR v == move v for this lane's entity, column == l15.
      const int Eo = 2 * t + half;
      const int bo = Eo / 19, eo = Eo % 19;
      if (eo < 18) {
        float s = 0.f;
#pragma unroll
        for (int v = 0; v < 8; ++v) s += acc[v];          // sum over 8 moves
        sm[OFF_MS + (bo * 18 + eo) * 16 + l15] = s;
      } else {                                            // active entity -> moves output
#pragma unroll
        for (int v = 0; v < 8; ++v)
          A.out[OUT_MOVES + (long long)(b0 + bo) * 128 + v * 16 + l15] = acc[v];
      }
    }
  }
  __syncthreads();

  // ============ Phase 1: gathers + enc_ms / slots / stats / boosts GEMMs ============
  // dex / item / ability gathers into dmia input (cols 0-7, 40-43, 44-51)
  for (int r = tid; r < NB * 18; r += 256) {
    const int b = r / 18, e = r % 18, gb = b0 + b;
    const int dex = (e < 12) ? A.public_dex[gb * 12 + e]     : A.private_dex[gb * 6 + e - 12];
    const int itm = (e < 12) ? A.public_item[gb * 12 + e]    : A.private_item[gb * 6 + e - 12];
    const int ab  = (e < 12) ? A.public_ability[gb * 12 + e] : A.private_ability[gb * 6 + e - 12];
    float* row = sm + OFF_DMIA + r * 52;
#pragma unroll
    for (int j = 0; j < 8; ++j) row[j]      = A.dex_tab[dex * 8 + j];
#pragma unroll
    for (int j = 0; j < 4; ++j) row[40 + j] = A.item_tab[itm * 4 + j];
#pragma unroll
    for (int j = 0; j < 8; ++j) row[44 + j] = A.ability_tab[ab * 8 + j];
  }
  // emb_weather (VALU, K=8 -> trivial) into field input cols 0-7 of both sides
  for (int t = tid; t < NB * 8; t += 256) {
    const int b = t >> 3, j = t & 7;
    float s = A.b_weather[j];
#pragma unroll
    for (int k = 0; k < 8; ++k) s += A.weather[(b0 + b) * 8 + k] * A.W_weather[j * 8 + k];
    sm[OFF_FIN + (b * 2)     * 24 + j] = s;
    sm[OFF_FIN + (b * 2 + 1) * 24 + j] = s;
  }
  // emb_slots: M=32 K=20 N=16 -> field input cols 8-23
  run_gemm<2, 1, 5>(
    [&](int r, int k0) { v2f x; const float* p = A.slot + (long long)(b0 * 2 + r) * 20;
      x.x = p[k0]; x.y = p[k0 + 1]; return x; },
    [&](int n, int k0) { v2f x; x.x = A.W_slot[n * 20 + k0]; x.y = A.W_slot[n * 20 + k0 + 1]; return x; },
    [&](int rt, int ct, v8f acc) {
      (void)ct;
#pragma unroll
      for (int v = 0; v < 8; ++v) { const int row = rt * 16 + (half ? v + 8 : v);
        sm[OFF_FIN + row * 24 + 8 + l15] = acc[v]; }
    });
  // emb_stats: M=96 K=6 N=32 -> oura cols 64-95 / ourb cols 64-95
  run_gemm<6, 2, 2>(
    [&](int r, int k0) { v2f x; const float* p = A.private_stats + (long long)(b0 * 6 + r) * 6;
      x.x = (k0 < 6) ? p[k0] : 0.f; x.y = (k0 + 1 < 6) ? p[k0 + 1] : 0.f; return x; },
    [&](int n, int k0) { v2f x; x.x = (k0 < 6) ? A.W_stat[n * 6 + k0] : 0.f;
      x.y = (k0 + 1 < 6) ? A.W_stat[n * 6 + k0 + 1] : 0.f; return x; },
    [&](int rt, int ct, v8f acc) {
#pragma unroll
      for (int v = 0; v < 8; ++v) {
        const int row = rt * 16 + (half ? v + 8 : v);
        const int b = row / 6, pe = row % 6, n = ct * 16 + l15;
        if (pe == 0) sm[OFF_OURA + b * 256 + 64 + n] = acc[v];
        else         sm[OFF_OURB + (b * 5 + pe - 1) * 192 + 64 + n] = acc[v];
      }
    });
  // emb_bv: M=32 K=171 N=48 -> bv buffer
  run_gemm<2, 3, 43>(
    [&](int r, int k0) { v2f x; const float* p = A.abv + (long long)(b0 * 2 + r) * 171;
      x.x = (k0 < 171) ? p[k0] : 0.f; x.y = (k0 + 1 < 171) ? p[k0 + 1] : 0.f; return x; },
    [&](int n, int k0) { v2f x; x.x = (k0 < 171) ? A.W_boosts[n * 171 + k0] : 0.f;
      x.y = (k0 + 1 < 171) ? A.W_boosts[n * 171 + k0 + 1] : 0.f; return x; },
    [&](int rt, int ct, v8f acc) {
#pragma unroll
      for (int v = 0; v < 8; ++v) { const int row = rt * 16 + (half ? v + 8 : v);
        sm[OFF_BV + row * 48 + ct * 16 + l15] = acc[v]; }
    });
  // enc_ms: M=288 K=16 N=32 (row-blocked, 2 col tiles in regs) -> dmia cols 8-39
  run_gemm_rb<2, 4>(18,
    [&](int r, int k0) { v2f x; x.x = sm[OFF_MS + r * 16 + k0]; x.y = sm[OFF_MS + r * 16 + k0 + 1]; return x; },
    [&](int n, int k0) { v2f x; x.x = A.W_moveset[n * 16 + k0]; x.y = A.W_moveset[n * 16 + k0 + 1]; return x; },
    [&](int rt, int ct, v8f acc) {
#pragma unroll
      for (int v = 0; v < 8; ++v) { const int row = rt * 16 + (half ? v + 8 : v);
        sm[OFF_DMIA + row * 52 + 8 + ct * 16 + l15] = acc[v]; }
    });
  __syncthreads();

  // ============ Phase 2: enc_dmia / field / enc_active GEMMs + hp/status gather ============
  // enc_dmia: M=288 K=52 N=64 (row-blocked, 4 col tiles in regs)
  run_gemm_rb<4, 13>(18,
    [&](int r, int k0) { v2f x; x.x = sm[OFF_DMIA + r * 52 + k0]; x.y = sm[OFF_DMIA + r * 52 + k0 + 1]; return x; },
    [&](int n, int k0) { v2f x; x.x = A.W_dmia[n * 52 + k0]; x.y = A.W_dmia[n * 52 + k0 + 1]; return x; },
    [&](int rt, int ct, v8f acc) {
#pragma unroll
      for (int v = 0; v < 8; ++v) {
        const int row = rt * 16 + (half ? v + 8 : v);
        const int b = row / 18, e = row % 18, n = ct * 16 + l15;
        if (e < 12)       sm[OFF_PUB + (b * 12 + e) * 72 + n] = acc[v];
        else if (e == 12) sm[OFF_OURA + b * 256 + 96 + n] = acc[v];
        else              sm[OFF_OURB + (b * 5 + e - 13) * 192 + 96 + n] = acc[v];
      }
    });
  // field: M=32 K=24 N=32 -> fld buffer
  run_gemm<2, 2, 6>(
    [&](int r, int k0) { v2f x; x.x = sm[OFF_FIN + r * 24 + k0]; x.y = sm[OFF_FIN + r * 24 + k0 + 1]; return x; },
    [&](int n, int k0) { v2f x; x.x = A.W_field[n * 24 + k0]; x.y = A.W_field[n * 24 + k0 + 1]; return x; },
    [&](int rt, int ct, v8f acc) {
#pragma unroll
      for (int v = 0; v < 8; ++v) { const int row = rt * 16 + (half ? v + 8 : v);
        sm[OFF_FLD + row * 32 + ct * 16 + l15] = acc[v]; }
    });
  // enc_active: M=32 K=48 N=64 -> oura cols 160-223 (side0), oppa cols 64-127 (side1)
  run_gemm<2, 4, 12>(
    [&](int r, int k0) { v2f x; x.x = sm[OFF_BV + r * 48 + k0]; x.y = sm[OFF_BV + r * 48 + k0 + 1]; return x; },
    [&](int n, int k0) { v2f x; x.x = A.W_active[n * 48 + k0]; x.y = A.W_active[n * 48 + k0 + 1]; return x; },
    [&](int rt, int ct, v8f acc) {
#pragma unroll
      for (int v = 0; v < 8; ++v) {
        const int row = rt * 16 + (half ? v + 8 : v);
        const int b = row / 2, sd = row % 2, n = ct * 16 + l15;
        if (sd == 0) sm[OFF_OURA + b * 256 + 160 + n] = acc[v];
        else         sm[OFF_OPPA + b * 160 + 64 + n]  = acc[v];
      }
    });
  // hp * W_hp.T and status_tab gather into pub_in cols 64-71
  for (int r = tid; r < NB * 12; r += 256) {
    const int b = r / 12, e = r % 12, gb = b0 + b;
    const float hp = A.public_hp[gb * 12 + e];
    const int st = A.public_status[gb * 12 + e];
    float* row = sm + OFF_PUB + r * 72;
#pragma unroll
    for (int j = 0; j < 4; ++j) row[64 + j] = hp * A.W_hp[j];
#pragma unroll
    for (int j = 0; j < 4; ++j) row[68 + j] = A.status_tab[st * 4 + j];
  }
  __syncthreads();

  // ============ Phase 3: enc_public GEMM + field broadcast into head inputs ============
  // enc_public: M=192 K=72 N=64 (row-blocked, 4 col tiles in regs)
  run_gemm_rb<4, 18>(12,
    [&](int r, int k0) { v2f x; x.x = sm[OFF_PUB + r * 72 + k0]; x.y = sm[OFF_PUB + r * 72 + k0 + 1]; return x; },
    [&](int n, int k0) { v2f x; x.x = A.W_public[n * 72 + k0]; x.y = A.W_public[n * 72 + k0 + 1]; return x; },
    [&](int rt, int ct, v8f acc) {
#pragma unroll
      for (int v = 0; v < 8; ++v) {
        const int row = rt * 16 + (half ? v + 8 : v);
        const int b = row / 12, e = row % 12, n = ct * 16 + l15;
        if (e == 0)      sm[OFF_OURA + b * 256 + n] = acc[v];
        else if (e < 6)  sm[OFF_OURB + (b * 5 + e - 1) * 192 + n] = acc[v];
        else if (e == 6) sm[OFF_OPPA + b * 160 + n] = acc[v];
        else             sm[OFF_OPPB + (b * 5 + e - 7) * 96 + n] = acc[v];
      }
    });
  // broadcast field embedding (repeat 6) into head input tails
  for (int t = tid; t < NB * 2 * 32; t += 256) {
    const int j = t & 31, sd = (t >> 5) & 1, b = t >> 6;
    const float v = sm[OFF_FLD + (b * 2 + sd) * 32 + j];
    if (sd == 0) {
      sm[OFF_OURA + b * 256 + 224 + j] = v;
#pragma unroll
      for (int e = 0; e < 5; ++e) sm[OFF_OURB + (b * 5 + e) * 192 + 160 + j] = v;
    } else {
      sm[OFF_OPPA + b * 160 + 128 + j] = v;
#pragma unroll
      for (int e = 0; e < 5; ++e) sm[OFF_OPPB + (b * 5 + e) * 96 + 64 + j] = v;
    }
  }
  __syncthreads();

  // ============ Phase 4: head GEMMs (+bias) -> d_out ============
  // our_active: M=16 K=256 N=128 (B fragments used once each; flat over 8 col tiles)
  run_gemm<1, 8, 64>(
    [&](int r, int k0) { v2f x; x.x = sm[OFF_OURA + r * 256 + k0]; x.y = sm[OFF_OURA + r * 256 + k0 + 1]; return x; },
    [&](int n, int k0) { v2f x; x.x = A.W_oura[n * 256 + k0]; x.y = A.W_oura[n * 256 + k0 + 1]; return x; },
    [&](int rt, int ct, v8f acc) {
      const int n = ct * 16 + l15; const float bias = A.b_oura[n];
#pragma unroll
      for (int v = 0; v < 8; ++v) { const int row = rt * 16 + (half ? v + 8 : v);
        A.out[(long long)(b0 + row) * 128 + n] = acc[v] + bias; }
    });
  // opp_active: M=16 K=160 N=128
  run_gemm<1, 8, 40>(
    [&](int r, int k0) { v2f x; x.x = sm[OFF_OPPA + r * 160 + k0]; x.y = sm[OFF_OPPA + r * 160 + k0 + 1]; return x; },
    [&](int n, int k0) { v2f x; x.x = A.W_oppa[n * 160 + k0]; x.y = A.W_oppa[n * 160 + k0 + 1]; return x; },
    [&](int rt, int ct, v8f acc) {
      const int n = ct * 16 + l15; const float bias = A.b_oppa[n];
#pragma unroll
      for (int v = 0; v < 8; ++v) { const int row = rt * 16 + (half ? v + 8 : v);
        A.out[OUT_OPPA + (long long)(b0 + row) * 128 + n] = acc[v] + bias; }
    });
  // Bench heads, column-blocked: waves 0-3 own our_bench col tiles (M=80 K=192 N=64),
  // waves 4-7 own opp_bench col tiles (M=80 K=96 N=64). Each streaming-weight B
  // fragment is loaded once and reused across all 5 row tiles (5x less L2 traffic).
  {
    const int wave = tid >> 5;
    if (wave < 4) {
      gemm_colblk<5, 48>(wave,
        [&](int r, int k0) { v2f x; x.x = sm[OFF_OURB + r * 192 + k0]; x.y = sm[OFF_OURB + r * 192 + k0 + 1]; return x; },
        [&](int n, int k0) { v2f x; x.x = A.W_ourb[n * 192 + k0]; x.y = A.W_ourb[n * 192 + k0 + 1]; return x; },
        [&](int rt, int ct, v8f acc) {
          const int n = ct * 16 + l15; const float bias = A.b_ourb[n];
#pragma unroll
          for (int v = 0; v < 8; ++v) { const int row = rt * 16 + (half ? v + 8 : v);
            A.out[OUT_OURB + (long long)(b0 * 5 + row) * 64 + n] = acc[v] + bias; }
        });
    } else {
      gemm_colblk<5, 24>(wave - 4,
        [&](int r, int k0) { v2f x; x.x = sm[OFF_OPPB + r * 96 + k0]; x.y = sm[OFF_OPPB + r * 96 + k0 + 1]; return x; },
        [&](int n, int k0) { v2f x; x.x = A.W_oppb[n * 96 + k0]; x.y = A.W_oppb[n * 96 + k0 + 1]; return x; },
        [&](int rt, int ct, v8f acc) {
          const int n = ct * 16 + l15; const float bias = A.b_oppb[n];
#pragma unroll
          for (int v = 0; v < 8; ++v) { const int row = rt * 16 + (half ? v + 8 : v);
            A.out[OUT_OPPB + (long long)(b0 * 5 + row) * 64 + n] = acc[v] + bias; }
        });
    }
  }
}

extern "C" void kernel_launch(void* const* d_in, const int* in_sizes, int n_in,
                              void* d_out, int out_size, void* d_ws, size_t ws_size,
                              hipStream_t stream) {
  (void)n_in; (void)out_size; (void)d_ws; (void)ws_size;
  Args a;
  a.public_dex      = (const int*)d_in[0];
  a.private_dex     = (const int*)d_in[1];
  a.public_moves    = (const int*)d_in[2];
  a.private_moves   = (const int*)d_in[3];
  a.active_moves    = (const int*)d_in[4];
  a.public_item     = (const int*)d_in[5];
  a.private_item    = (const int*)d_in[6];
  a.public_ability  = (const int*)d_in[7];
  a.private_ability = (const int*)d_in[8];
  a.public_hp       = (const float*)d_in[9];
  a.public_status   = (const int*)d_in[10];
  a.private_stats   = (const float*)d_in[11];
  a.weather         = (const float*)d_in[12];
  a.slot            = (const float*)d_in[13];
  a.abv             = (const float*)d_in[14];
  a.dex_tab         = (const float*)d_in[15];
  a.move_id_tab     = (const float*)d_in[16];
  a.move_pp_tab     = (const float*)d_in[17];
  a.item_tab        = (const float*)d_in[18];
  a.ability_tab     = (const float*)d_in[19];
  a.status_tab      = (const float*)d_in[20];
  a.W_move          = (const float*)d_in[21];
  a.b_move          = (const float*)d_in[22];
  a.W_moveset       = (const float*)d_in[23];
  a.W_hp            = (const float*)d_in[24];
  a.W_stat          = (const float*)d_in[25];
  a.W_dmia          = (const float*)d_in[26];
  a.W_weather       = (const float*)d_in[27];
  a.b_weather       = (const float*)d_in[28];
  a.W_slot          = (const float*)d_in[29];
  a.W_boosts        = (const float*)d_in[30];
  a.W_public        = (const float*)d_in[31];
  a.W_active        = (const float*)d_in[32];
  a.W_field         = (const float*)d_in[33];
  a.W_oura          = (const float*)d_in[34];
  a.b_oura          = (const float*)d_in[35];
  a.W_ourb          = (const float*)d_in[36];
  a.b_ourb          = (const float*)d_in[37];
  a.W_oppa          = (const float*)d_in[38];
  a.b_oppa          = (const float*)d_in[39];
  a.W_oppb          = (const float*)d_in[40];
  a.b_oppb          = (const float*)d_in[41];
  a.out             = (float*)d_out;

  const int Btot = in_sizes[0] / 12;   // public_dex is (B,12,1)
  (void)hipFuncSetAttribute((const void*)obs_encoder_kernel,
                            hipFuncAttributeMaxDynamicSharedMemorySize, LDS_BYTES);
  obs_encoder_kernel<<<Btot / NB, 256, LDS_BYTES, stream>>>(a, Btot);
}